// LIF_layer_43207370998031
// MI455X (gfx1250) — compile-verified
//
#include <hip/hip_runtime.h>

// LIF scan: v = v*DECAY + I[t]; spike = (v - THR >= 0); v -= spike*THR
// Shapes: T=16, N = B*C*H*W = 4,194,304. Input/output layout (T, N), time outermost.
// Memory-bound streaming kernel: 512 MB total traffic -> ~22 us floor at 23.3 TB/s.

#define LIF_T     16
#define V_DECAY   0.5f
#define V_THR     1.0f

typedef float v4f __attribute__((ext_vector_type(4)));

__global__ __launch_bounds__(256) void lif_scan_kernel(const float* __restrict__ in,
                                                       float* __restrict__ out,
                                                       int nvec /* = N/4 */) {
    int i = blockIdx.x * blockDim.x + threadIdx.x;
    if (i >= nvec) return;

    const v4f* __restrict__ pin  = (const v4f*)in;
    v4f*       __restrict__ pout = (v4f*)out;

    v4f v = {0.0f, 0.0f, 0.0f, 0.0f};   // V_INIT = 0

    // Fully unrolled over time. Loads are independent of the serial v-chain,
    // so the compiler can issue them ahead (MLP) and wait incrementally.
    // Non-temporal hints: every byte is touched exactly once -> don't pollute L2.
#pragma unroll
    for (int t = 0; t < LIF_T; ++t) {
        size_t off = (size_t)t * (size_t)nvec + (size_t)i;
        v4f x = __builtin_nontemporal_load(pin + off);

        // integrate with decay
        v = v * V_DECAY + x;

        // spike = (v - thr >= 0) as float; soft (subtract) reset
        v4f s;
#pragma unroll
        for (int k = 0; k < 4; ++k) {
            s[k] = ((v[k] - V_THR) >= 0.0f) ? 1.0f : 0.0f;
        }
        v = v - s * V_THR;

        __builtin_nontemporal_store(s, pout + off);
    }
}

extern "C" void kernel_launch(void* const* d_in, const int* in_sizes, int n_in,
                              void* d_out, int out_size, void* d_ws, size_t ws_size,
                              hipStream_t stream) {
    const float* in  = (const float*)d_in[0];
    float*       out = (float*)d_out;

    const int total = in_sizes[0];        // T * B * C * H * W = 67,108,864
    const int n     = total / LIF_T;      // elements per time step = 4,194,304
    const int nvec  = n / 4;              // float4 vectors per time step = 1,048,576

    const int block = 256;                // 8 wave32 waves
    const int grid  = (nvec + block - 1) / block;  // 4096 blocks

    lif_scan_kernel<<<grid, block, 0, stream>>>(in, out, nvec);
}